// SSTInputLayer_37701222924672
// MI455X (gfx1250) — compile-verified
//
#include <hip/hip_runtime.h>
#include <hip/hip_bf16.h>

// ---- static config (mirrors reference) ----
#define C_DIM   128
#define WXY     12
#define NWXY    40
#define NWIN    6400          // B(4) * 40 * 40
#define MAX_T   36
#define NHIST   (NWIN + 1)    // extra "dummy" segment for dropped/invalid voxels
#define CHUNK   1024
#define NSLOTS  (NWIN * MAX_T)

typedef __attribute__((ext_vector_type(2))) float v2f;
typedef __attribute__((ext_vector_type(8))) float v8f;

// ---------------- window ids ----------------
__global__ void k_wins(const int* __restrict__ coors, int* __restrict__ win0,
                       int* __restrict__ win1, int n) {
    int i = blockIdx.x * blockDim.x + threadIdx.x;
    if (i >= n) return;
    int b = coors[i * 4 + 0];
    int y = coors[i * 4 + 2];
    int x = coors[i * 4 + 3];
    win0[i] = b * (NWXY * NWXY) + (y / WXY) * NWXY + (x / WXY);
    int sy = y + (WXY / 2), sx = x + (WXY / 2);
    win1[i] = b * (NWXY * NWXY) + (sy / WXY) * NWXY + (sx / WXY);
}

// ---------------- rank phase 1: per-chunk histogram (LDS, 6401 bins) ----------------
__global__ void __launch_bounds__(CHUNK)
k_hist(const int* __restrict__ win, const int* __restrict__ keep, int useKeep,
       int* __restrict__ hist, int n) {
    __shared__ int sh[NHIST];
    for (int t = threadIdx.x; t < NHIST; t += blockDim.x) sh[t] = 0;
    __syncthreads();
    int i = blockIdx.x * CHUNK + threadIdx.x;
    if (i < n) {
        int w = win[i];
        if (useKeep && !keep[i]) w = NWIN;
        atomicAdd(&sh[w], 1);
    }
    __syncthreads();
    int* row = hist + (size_t)blockIdx.x * NHIST;
    for (int t = threadIdx.x; t < NHIST; t += blockDim.x) row[t] = sh[t];
}

// ---------------- rank phase 2: exclusive scan over chunks, per window ----------------
__global__ void k_scan(int* __restrict__ hist, int nchunk) {
    int t = blockIdx.x * blockDim.x + threadIdx.x;
    if (t >= NHIST) return;
    int run = 0;
    for (int c = 0; c < nchunk; ++c) {
        int* p = hist + (size_t)c * NHIST + t;
        int v = *p;
        *p = run;
        run += v;
    }
}

// ---------------- rank phase 3: chunk-local rank + global offset ----------------
// Local scan vectorized as int4 (ds_load_b128) to cut DS instruction count 4x.
__global__ void __launch_bounds__(CHUNK)
k_rank(const int* __restrict__ win, const int* __restrict__ keep, int useKeep,
       const int* __restrict__ hist, int* __restrict__ rank, int n) {
    __shared__ int sh[CHUNK];
    int i = blockIdx.x * CHUNK + threadIdx.x;
    int w = NWIN;
    if (i < n) {
        w = win[i];
        if (useKeep && !keep[i]) w = NWIN;
    }
    int tid = threadIdx.x;
    sh[tid] = w;
    __syncthreads();
    if (i >= n) return;
    int cnt = 0;
    const int4* sh4 = (const int4*)sh;
    int nfull = tid >> 2;
    for (int q = 0; q < nfull; ++q) {
        int4 v = sh4[q];
        cnt += (v.x == w) ? 1 : 0;
        cnt += (v.y == w) ? 1 : 0;
        cnt += (v.z == w) ? 1 : 0;
        cnt += (v.w == w) ? 1 : 0;
    }
    for (int j = nfull << 2; j < tid; ++j) cnt += (sh[j] == w) ? 1 : 0;
    rank[i] = hist[(size_t)blockIdx.x * NHIST + w] + cnt;
}

// ---------------- keep-mask updates ----------------
__global__ void k_keep(const int* __restrict__ rank, int* __restrict__ keep,
                       int mode, int n) {
    int i = blockIdx.x * blockDim.x + threadIdx.x;
    if (i >= n) return;
    int ok = rank[i] < MAX_T;
    keep[i] = (mode == 0) ? ok : (keep[i] && ok);
}

// ---------------- inverse map: slot -> voxel ----------------
__global__ void k_fillmap(int* __restrict__ m, int n) {
    int i = blockIdx.x * blockDim.x + threadIdx.x;
    if (i < n) m[i] = -1;
}

__global__ void k_scatter(const int* __restrict__ win, const int* __restrict__ keep,
                          const int* __restrict__ rank, int* __restrict__ map, int n) {
    int i = blockIdx.x * blockDim.x + threadIdx.x;
    if (i >= n) return;
    if (!keep[i]) return;
    int r = rank[i];
    if (r < MAX_T) map[win[i] * MAX_T + r] = i;
}

// ---------------- position embedding + mask (each output written exactly once) ----------------
__global__ void __launch_bounds__(C_DIM)
k_pe(const int* __restrict__ map, const int* __restrict__ coors,
     float* __restrict__ pe, float* __restrict__ mask, int shift) {
    int slot = blockIdx.x;            // [0, NWIN*MAX_T)
    int c = threadIdx.x;              // [0, 128)
    int v = map[slot];
    float val = 0.0f;
    if (v >= 0) {
        int y = coors[v * 4 + 2] + shift;
        int x = coors[v * 4 + 3] + shift;
        float coord = (c < C_DIM / 2) ? ((float)(x % WXY) - (float)WXY * 0.5f)
                                      : ((float)(y % WXY) - (float)WXY * 0.5f);
        int k = c & (C_DIM / 2 - 1);
        float e = (float)(2 * (k >> 1)) * (1.0f / (float)(C_DIM / 2));
        // inv_freq = 10000^e  ->  exp2(log2(10000) * e)
        float inv_freq = exp2f(13.2877123795494f * e);
        float arg = coord / inv_freq;
        val = ((k & 1) == 0) ? sinf(arg) : cosf(arg);
    }
    pe[(size_t)slot * C_DIM + c] = val;
    if (c == 0) mask[slot] = (v < 0) ? 1.0f : 0.0f;
}

// ---------------- voxel_out = diag(keep) x feats via V_WMMA_F32_16X16X4_F32 ----------------
// One wave handles a 16-voxel x 128-feature tile: 8 column tiles x 4 K-chunk WMMAs.
// Multiplies are by exactly 1.0/0.0 and accumulations add 0.0, so result is f32-exact.
// Fast path for full tiles is branch-free (EXEC stays all-ones, no saveexec churn).
__global__ void __launch_bounds__(256)
k_vout(const float* __restrict__ feats, const int* __restrict__ keep,
       float* __restrict__ out, int n) {
    int wave = (int)((blockIdx.x * blockDim.x + threadIdx.x) >> 5);
    int lane = threadIdx.x & 31;
    int ntiles = (n + 15) >> 4;
    if (wave >= ntiles) return;           // wave-uniform
    int row0 = wave << 4;
    int m = lane & 15;                    // M index (lanes 0-15) / column index for B
    bool hi = lane >= 16;

    if (row0 + 16 <= n) {
        // -------- full tile: no guards anywhere --------
        float kv = keep[row0 + m] ? 1.0f : 0.0f;
        for (int ct = 0; ct < 8; ++ct) {
            int c0 = ct * 16;
            v8f acc = {};
#pragma unroll
            for (int j = 0; j < 4; ++j) {
                int k0 = 4 * j + (hi ? 2 : 0);
                v2f A;
                A.x = (m == k0) ? kv : 0.0f;
                A.y = (m == (k0 + 1)) ? kv : 0.0f;
                v2f B;
                B.x = feats[(size_t)(row0 + k0) * C_DIM + c0 + m];
                B.y = feats[(size_t)(row0 + k0 + 1) * C_DIM + c0 + m];
                acc = __builtin_amdgcn_wmma_f32_16x16x4_f32(
                    false, A, false, B, (short)0, acc, false, false);
            }
#pragma unroll
            for (int r = 0; r < 8; ++r) {
                int orow = row0 + r + (hi ? 8 : 0);
                out[(size_t)orow * C_DIM + c0 + m] = acc[r];
            }
        }
    } else {
        // -------- last partial tile: clamped loads, guarded stores --------
        int rm = row0 + m;
        int rmc = rm < n ? rm : (n - 1);
        float kv = ((rm < n) && keep[rmc]) ? 1.0f : 0.0f;
        for (int ct = 0; ct < 8; ++ct) {
            int c0 = ct * 16;
            v8f acc = {};
#pragma unroll
            for (int j = 0; j < 4; ++j) {
                int k0 = 4 * j + (hi ? 2 : 0);
                v2f A;
                A.x = (m == k0) ? kv : 0.0f;
                A.y = (m == (k0 + 1)) ? kv : 0.0f;
                int r0 = row0 + k0;
                int r1 = r0 + 1;
                int cr0 = r0 < n ? r0 : (n - 1);
                int cr1 = r1 < n ? r1 : (n - 1);
                float b0 = feats[(size_t)cr0 * C_DIM + c0 + m];
                float b1 = feats[(size_t)cr1 * C_DIM + c0 + m];
                v2f B;
                B.x = (r0 < n) ? b0 : 0.0f;
                B.y = (r1 < n) ? b1 : 0.0f;
                acc = __builtin_amdgcn_wmma_f32_16x16x4_f32(
                    false, A, false, B, (short)0, acc, false, false);
            }
#pragma unroll
            for (int r = 0; r < 8; ++r) {
                int orow = row0 + r + (hi ? 8 : 0);
                if (orow < n) out[(size_t)orow * C_DIM + c0 + m] = acc[r];
            }
        }
    }
}

// ---------------- host launch ----------------
static inline size_t align256(size_t x) { return (x + 255) & ~(size_t)255; }

extern "C" void kernel_launch(void* const* d_in, const int* in_sizes, int n_in,
                              void* d_out, int out_size, void* d_ws, size_t ws_size,
                              hipStream_t stream) {
    (void)n_in; (void)out_size; (void)ws_size;
    const float* feats = (const float*)d_in[0];
    const int* coors = (const int*)d_in[1];
    const int N = in_sizes[1] / 4;
    const int nchunk = (N + CHUNK - 1) / CHUNK;

    // output layout: voxel_out | pe0 | pe1 | mask0 | mask1 (flat, return order)
    float* out = (float*)d_out;
    float* voxel_out = out;
    float* pe0 = voxel_out + (size_t)N * C_DIM;
    float* pe1 = pe0 + (size_t)NSLOTS * C_DIM;
    float* mask0 = pe1 + (size_t)NSLOTS * C_DIM;
    float* mask1 = mask0 + (size_t)NSLOTS;

    // scratch layout
    char* ws = (char*)d_ws;
    size_t off = 0;
    int* win0 = (int*)(ws + off); off += align256((size_t)N * 4);
    int* win1 = (int*)(ws + off); off += align256((size_t)N * 4);
    int* rank = (int*)(ws + off); off += align256((size_t)N * 4);
    int* keep = (int*)(ws + off); off += align256((size_t)N * 4);
    int* map0 = (int*)(ws + off); off += align256((size_t)NSLOTS * 4);
    int* map1 = (int*)(ws + off); off += align256((size_t)NSLOTS * 4);
    int* hist = (int*)(ws + off); off += align256((size_t)nchunk * NHIST * 4);

    const int TB = 256;
    const int gN = (N + TB - 1) / TB;
    const int gS = (NSLOTS + TB - 1) / TB;
    const int gH = (NHIST + TB - 1) / TB;

    k_wins<<<gN, TB, 0, stream>>>(coors, win0, win1, N);

    // Pass A: r0 over win0 (all voxels) -> keep0
    k_hist<<<nchunk, CHUNK, 0, stream>>>(win0, keep, 0, hist, N);
    k_scan<<<gH, TB, 0, stream>>>(hist, nchunk);
    k_rank<<<nchunk, CHUNK, 0, stream>>>(win0, keep, 0, hist, rank, N);
    k_keep<<<gN, TB, 0, stream>>>(rank, keep, 0, N);

    // Pass B: r1 over win1 among keep0 -> keep
    k_hist<<<nchunk, CHUNK, 0, stream>>>(win1, keep, 1, hist, N);
    k_scan<<<gH, TB, 0, stream>>>(hist, nchunk);
    k_rank<<<nchunk, CHUNK, 0, stream>>>(win1, keep, 1, hist, rank, N);
    k_keep<<<gN, TB, 0, stream>>>(rank, keep, 1, N);

    // Pass C: r0f among keep -> map0
    k_hist<<<nchunk, CHUNK, 0, stream>>>(win0, keep, 1, hist, N);
    k_scan<<<gH, TB, 0, stream>>>(hist, nchunk);
    k_rank<<<nchunk, CHUNK, 0, stream>>>(win0, keep, 1, hist, rank, N);
    k_fillmap<<<gS, TB, 0, stream>>>(map0, NSLOTS);
    k_scatter<<<gN, TB, 0, stream>>>(win0, keep, rank, map0, N);

    // Pass D: r1f among keep -> map1
    k_hist<<<nchunk, CHUNK, 0, stream>>>(win1, keep, 1, hist, N);
    k_scan<<<gH, TB, 0, stream>>>(hist, nchunk);
    k_rank<<<nchunk, CHUNK, 0, stream>>>(win1, keep, 1, hist, rank, N);
    k_fillmap<<<gS, TB, 0, stream>>>(map1, NSLOTS);
    k_scatter<<<gN, TB, 0, stream>>>(win1, keep, rank, map1, N);

    // Dense outputs: each element written exactly once
    k_pe<<<NSLOTS, C_DIM, 0, stream>>>(map0, coors, pe0, mask0, 0);
    k_pe<<<NSLOTS, C_DIM, 0, stream>>>(map1, coors, pe1, mask1, WXY / 2);

    // voxel_out via WMMA (16 voxels per wave, 8 waves per block)
    int ntiles = (N + 15) / 16;
    int nthreads = ntiles * 32;
    k_vout<<<(nthreads + 255) / 256, 256, 0, stream>>>(feats, keep, voxel_out, N);
}